// AdditiveAttention_5145370821130
// MI455X (gfx1250) — compile-verified
//
#include <hip/hip_runtime.h>

typedef __attribute__((ext_vector_type(2))) float v2f;
typedef __attribute__((ext_vector_type(8))) float v8f;

#if __has_builtin(__builtin_amdgcn_tanhf)
#define DEV_TANH(x) __builtin_amdgcn_tanhf(x)
#else
#define DEV_TANH(x) tanhf(x)
#endif

// ---------------------------------------------------------------------------
// Generic (optionally batched) row-major GEMM: C[M,N] = A[M,Kd] * W[Kd,N]
// One wave (32 threads) computes one 16x16 output tile via V_WMMA_F32_16X16X4_F32.
// grid = (M/16, N/16, nbatch); batch strides in elements.
// ---------------------------------------------------------------------------
__global__ __launch_bounds__(32)
void gemm_f32_wmma(const float* __restrict__ A, const float* __restrict__ W,
                   float* __restrict__ C, int Kd, int N,
                   long batchA, long batchB, long batchC) {
    const int mt = blockIdx.x, nt = blockIdx.y, bz = blockIdx.z;
    A += (size_t)bz * batchA;
    W += (size_t)bz * batchB;
    C += (size_t)bz * batchC;

    const int lane = threadIdx.x;       // 0..31
    const int half = lane >> 4;         // 0 | 1 -> selects K pair {0,1} vs {2,3}
    const int mr   = lane & 15;         // M index for A operand, N index for B/C

    const float* arow = A + (size_t)(mt * 16 + mr) * Kd + half * 2;
    const float* wcol = W + (size_t)(half * 2) * N + nt * 16 + mr;

    v8f acc = {};
    for (int kk = 0; kk < Kd; kk += 4) {
        v2f a;
        a.x = arow[kk + 0];
        a.y = arow[kk + 1];
        v2f b;
        b.x = wcol[(size_t)kk * N];
        b.y = wcol[(size_t)kk * N + N];
        // (neg_a, A, neg_b, B, c_mod, C, reuse_a, reuse_b)
        acc = __builtin_amdgcn_wmma_f32_16x16x4_f32(
            false, a, false, b, (short)0, acc, false, false);
    }

    // D layout: VGPR j holds M=j (lanes 0-15) / M=8+j (lanes 16-31), N = lane&15
    float* crow = C + (size_t)(mt * 16 + half * 8) * N + nt * 16 + mr;
#pragma unroll
    for (int j = 0; j < 8; ++j) crow[(size_t)j * N] = acc[j];
}

// ---------------------------------------------------------------------------
// Fused scores + masked softmax.
// One 256-thread block per (b,q) row. Each wave processes 4 k values at a
// time: lanes are split into 4 groups of 8; each lane owns 16 consecutive h
// (4x float4 loads), so the H=128 reduction needs only a 3-step shfl_xor
// (masks 1,2,4 stay inside each 8-lane group; all 4 groups reduce in
// parallel) -> ~0.05 DS ops per tanh instead of 1.25.
// Then block-wide masked softmax over K=1024 in LDS.
// ---------------------------------------------------------------------------
__global__ __launch_bounds__(256)
void scores_softmax_kernel(const float* __restrict__ qp,   // [B,Q,H]
                           const float* __restrict__ kp,   // [B,K,H]
                           const float* __restrict__ wv,   // [H]
                           const int*   __restrict__ vlen, // [B]
                           float* __restrict__ attn) {     // [B,Q,K]
    const int q = blockIdx.x;          // 0..63
    const int b = blockIdx.y;          // 0..15
    const int t = threadIdx.x;         // 0..255
    const int wave = t >> 5;
    const int lane = t & 31;
    const int ksub = lane >> 3;        // 0..3  : which of 4 concurrent k
    const int hq   = lane & 7;         // 0..7  : h-chunk within the k-group
    const int h0   = hq * 16;          // 16 h values per lane

    __shared__ float sc[1024];
    __shared__ float redm[8];
    __shared__ float reds[8];

    const float* qrow = qp + ((size_t)(b * 64 + q)) * 128 + h0;
    const float4 qv0 = *(const float4*)(qrow + 0);
    const float4 qv1 = *(const float4*)(qrow + 4);
    const float4 qv2 = *(const float4*)(qrow + 8);
    const float4 qv3 = *(const float4*)(qrow + 12);
    const float4 wv0 = *(const float4*)(wv + h0 + 0);
    const float4 wv1 = *(const float4*)(wv + h0 + 4);
    const float4 wv2 = *(const float4*)(wv + h0 + 8);
    const float4 wv3 = *(const float4*)(wv + h0 + 12);
    const int vl = vlen[b];
    const float* kpb = kp + (size_t)b * 1024 * 128;

    const int k0 = wave * 128;
    for (int k4 = k0; k4 < k0 + 128; k4 += 4) {
        const int k = k4 + ksub;
        const float* kr = kpb + (size_t)k * 128 + h0;
        const float4 kv0 = *(const float4*)(kr + 0);
        const float4 kv1 = *(const float4*)(kr + 4);
        const float4 kv2 = *(const float4*)(kr + 8);
        const float4 kv3 = *(const float4*)(kr + 12);

        float s = wv0.x * DEV_TANH(qv0.x + kv0.x);
        s = fmaf(wv0.y, DEV_TANH(qv0.y + kv0.y), s);
        s = fmaf(wv0.z, DEV_TANH(qv0.z + kv0.z), s);
        s = fmaf(wv0.w, DEV_TANH(qv0.w + kv0.w), s);
        s = fmaf(wv1.x, DEV_TANH(qv1.x + kv1.x), s);
        s = fmaf(wv1.y, DEV_TANH(qv1.y + kv1.y), s);
        s = fmaf(wv1.z, DEV_TANH(qv1.z + kv1.z), s);
        s = fmaf(wv1.w, DEV_TANH(qv1.w + kv1.w), s);
        s = fmaf(wv2.x, DEV_TANH(qv2.x + kv2.x), s);
        s = fmaf(wv2.y, DEV_TANH(qv2.y + kv2.y), s);
        s = fmaf(wv2.z, DEV_TANH(qv2.z + kv2.z), s);
        s = fmaf(wv2.w, DEV_TANH(qv2.w + kv2.w), s);
        s = fmaf(wv3.x, DEV_TANH(qv3.x + kv3.x), s);
        s = fmaf(wv3.y, DEV_TANH(qv3.y + kv3.y), s);
        s = fmaf(wv3.z, DEV_TANH(qv3.z + kv3.z), s);
        s = fmaf(wv3.w, DEV_TANH(qv3.w + kv3.w), s);

        // reduce across the 8-lane h-group (4 k-groups reduce in parallel)
        s += __shfl_xor(s, 1, 32);
        s += __shfl_xor(s, 2, 32);
        s += __shfl_xor(s, 4, 32);
        if (hq == 0) sc[k] = (k < vl) ? s : -1e6f;
    }
    __syncthreads();

    // ---- block softmax over sc[0..1023] ----
    const float v0 = sc[t], v1 = sc[t + 256], v2 = sc[t + 512], v3 = sc[t + 768];
    float m = fmaxf(fmaxf(v0, v1), fmaxf(v2, v3));
#pragma unroll
    for (int off = 16; off > 0; off >>= 1) m = fmaxf(m, __shfl_xor(m, off, 32));
    if (lane == 0) redm[wave] = m;
    __syncthreads();
    float bm = redm[0];
#pragma unroll
    for (int i = 1; i < 8; ++i) bm = fmaxf(bm, redm[i]);

    const float e0 = expf(v0 - bm), e1 = expf(v1 - bm);
    const float e2 = expf(v2 - bm), e3 = expf(v3 - bm);
    float ssum = (e0 + e1) + (e2 + e3);
#pragma unroll
    for (int off = 16; off > 0; off >>= 1) ssum += __shfl_xor(ssum, off, 32);
    if (lane == 0) reds[wave] = ssum;
    __syncthreads();
    float tot = reds[0];
#pragma unroll
    for (int i = 1; i < 8; ++i) tot += reds[i];
    const float inv = 1.0f / tot;

    float* arow = attn + ((size_t)(b * 64 + q)) * 1024;
    arow[t]       = e0 * inv;
    arow[t + 256] = e1 * inv;
    arow[t + 512] = e2 * inv;
    arow[t + 768] = e3 * inv;
}

// ---------------------------------------------------------------------------
// Launch: qproj GEMM, kproj GEMM, fused scores+softmax, batched attn@values.
// ---------------------------------------------------------------------------
extern "C" void kernel_launch(void* const* d_in, const int* in_sizes, int n_in,
                              void* d_out, int out_size, void* d_ws, size_t ws_size,
                              hipStream_t stream) {
    const float* queries = (const float*)d_in[0];  // [16,64,256]
    const float* keys    = (const float*)d_in[1];  // [16,1024,256]
    const float* values  = (const float*)d_in[2];  // [16,1024,256]
    const int*   valid   = (const int*)  d_in[3];  // [16]
    const float* Wq      = (const float*)d_in[4];  // [256,128]
    const float* Wk      = (const float*)d_in[5];  // [256,128]
    const float* wv      = (const float*)d_in[6];  // [128]
    float* out = (float*)d_out;                    // [16,64,256]

    float* qproj = (float*)d_ws;                       // 1024*128   (0.5 MB)
    float* kproj = qproj + (size_t)1024 * 128;         // 16384*128  (8 MB)
    float* attn  = kproj + (size_t)16384 * 128;        // 1024*1024  (4 MB)

    // qproj[1024,128] = queries[1024,256] @ Wq[256,128]
    gemm_f32_wmma<<<dim3(1024 / 16, 128 / 16, 1), 32, 0, stream>>>(
        queries, Wq, qproj, 256, 128, 0, 0, 0);

    // kproj[16384,128] = keys[16384,256] @ Wk[256,128]
    gemm_f32_wmma<<<dim3(16384 / 16, 128 / 16, 1), 32, 0, stream>>>(
        keys, Wk, kproj, 256, 128, 0, 0, 0);

    // attn[B,Q,K]: scores + masked softmax (dominant tanh stage)
    scores_softmax_kernel<<<dim3(64, 16), 256, 0, stream>>>(
        qproj, kproj, wv, valid, attn);

    // out[b] (64x256) = attn[b] (64x1024) @ values[b] (1024x256), b = 0..15
    gemm_f32_wmma<<<dim3(64 / 16, 256 / 16, 16), 32, 0, stream>>>(
        attn, values, out, 1024, 256,
        (long)64 * 1024, (long)1024 * 256, (long)64 * 256);
}